// SCRFDPostproc_52707838656734
// MI455X (gfx1250) — compile-verified
//
#include <hip/hip_runtime.h>
#include <stdint.h>

#ifndef __has_builtin
#define __has_builtin(x) 0
#endif

#define THRD 256

constexpr int   kN8   = 640 * 640 * 2;   // 819200
constexpr int   kN16  = 320 * 320 * 2;   // 204800
constexpr int   kN32  = 160 * 160 * 2;   // 51200
constexpr int   kNTot = kN8 + kN16 + kN32;
constexpr int   kK    = 2048;
constexpr int   kCap  = 4096;
constexpr int   kBins = 4096;
constexpr float kThr  = 0.5f;
constexpr float kNmsT = 0.4f;
constexpr int   kTile = 1024;            // floats per block tile (256 thr * 4)

// ---- workspace layout (bytes) ----
constexpr size_t WS_HIST = 0;                                    // u32[kBins]
constexpr size_t WS_CNT  = WS_HIST + (size_t)kBins * 4;          // u32
constexpr size_t WS_CUT  = WS_CNT + 4;                           // u32
constexpr size_t WS_CAND = (WS_CUT + 4 + 7) & ~(size_t)7;        // u64[kCap]
constexpr size_t WS_DEC  = WS_CAND + (size_t)kCap * 8;           // f32[kK*16]
constexpr size_t WS_VAL  = WS_DEC + (size_t)kK * 16 * 4;         // u32[kK]
constexpr size_t WS_KEEP = WS_VAL + (size_t)kK * 4;              // u32[64]
constexpr size_t WS_SUPP = WS_KEEP + 64 * 4;                     // u32[kK*64]

// ---- gfx1250 async global->LDS path ----
#if __has_builtin(__builtin_amdgcn_global_load_async_to_lds_b128)
#define HAVE_ASYNC_LDS 1
#endif

typedef int v4i __attribute__((vector_size(16)));
typedef __attribute__((address_space(1))) v4i gas_v4i;
typedef __attribute__((address_space(3))) v4i las_v4i;

__device__ __forceinline__ void asyncCopy16(const float* g, float* l) {
#if defined(HAVE_ASYNC_LDS)
  __builtin_amdgcn_global_load_async_to_lds_b128((gas_v4i*)(uintptr_t)g,
                                                 (las_v4i*)(uintptr_t)l, 0, 0);
#else
  (void)g; (void)l;
#endif
}

__device__ __forceinline__ void waitAsync0() {
#if __has_builtin(__builtin_amdgcn_s_wait_asynccnt)
  __builtin_amdgcn_s_wait_asynccnt(0);
#else
  asm volatile("s_wait_asynccnt 0x0" ::: "memory");
#endif
}
__device__ __forceinline__ void waitAsync1() {
#if __has_builtin(__builtin_amdgcn_s_wait_asynccnt)
  __builtin_amdgcn_s_wait_asynccnt(1);
#else
  asm volatile("s_wait_asynccnt 0x1" ::: "memory");
#endif
}

__device__ __forceinline__ void histAdd(unsigned int* lhist, float x) {
  if (x >= kThr) {
    int b = (int)((x - kThr) * ((float)kBins / kThr));
    b = min(max(b, 0), kBins - 1);
    atomicAdd(&lhist[b], 1u);
  }
}

// ---- kernel 1 (defined first so --disasm snippet shows the async pipeline):
// histogram of scores, async global->LDS double-buffered stream ----
__global__ __launch_bounds__(THRD) void k_hist(const float* __restrict__ s, int n,
                                               unsigned int* __restrict__ hist) {
  __shared__ unsigned int lhist[kBins];
  const int tid = threadIdx.x;
  for (int b = tid; b < kBins; b += THRD) lhist[b] = 0u;
  __syncthreads();

  const int ntiles = (n + kTile - 1) / kTile;

#if defined(HAVE_ASYNC_LDS)
  __shared__ float buf[2][kTile];
  int cur = blockIdx.x;
  const int step = gridDim.x;
  int bsel = 0;
  bool curFull = (cur < ntiles) && ((cur + 1) * kTile <= n);
  if (curFull) asyncCopy16(s + (size_t)cur * kTile + tid * 4, &buf[0][tid * 4]);
  while (cur < ntiles) {
    const int nxt = cur + step;
    const bool nxtFull = (nxt < ntiles) && ((nxt + 1) * kTile <= n);
    if (nxtFull) asyncCopy16(s + (size_t)nxt * kTile + tid * 4, &buf[bsel ^ 1][tid * 4]);
    if (curFull) {
      if (nxtFull) waitAsync1(); else waitAsync0();
      const float4 v = *(const float4*)&buf[bsel][tid * 4];
      histAdd(lhist, v.x); histAdd(lhist, v.y);
      histAdd(lhist, v.z); histAdd(lhist, v.w);
    } else {
      const int base = cur * kTile + tid * 4;
      #pragma unroll
      for (int c = 0; c < 4; ++c)
        if (base + c < n) histAdd(lhist, s[base + c]);
    }
    cur = nxt; curFull = nxtFull; bsel ^= 1;
  }
#else
  for (int i = blockIdx.x * THRD + tid; i < n; i += gridDim.x * THRD)
    histAdd(lhist, s[i]);
#endif

  __syncthreads();
  for (int b = tid; b < kBins; b += THRD) {
    unsigned int c = lhist[b];
    if (c) atomicAdd(&hist[b], c);
  }
}

// ---- kernel 0: zero the workspace pieces we read-modify ----
__global__ __launch_bounds__(1024) void k_init(unsigned int* hist,
                                               unsigned int* cnt,
                                               unsigned long long* cand) {
  int t = threadIdx.x;
  for (int i = t; i < kBins; i += 1024) hist[i] = 0u;
  for (int i = t; i < kCap; i += 1024) cand[i] = 0ull;
  if (t == 0) *cnt = 0u;
}

// ---- kernel 2: suffix scan -> cutoff bin ----
__global__ void k_scan(const unsigned int* __restrict__ hist,
                       unsigned int* __restrict__ cut) {
  if (threadIdx.x == 0 && blockIdx.x == 0) {
    unsigned int acc = 0;
    int c = 0;
    for (int b = kBins - 1; b >= 0; --b) {
      acc += hist[b];
      if (acc >= (unsigned)kK) { c = b; break; }
    }
    *cut = (unsigned int)c;
  }
}

// ---- kernel 3: collect candidates >= cutoff bin as sortable u64 keys ----
__global__ __launch_bounds__(THRD) void k_collect(const float* __restrict__ s, int n,
                                                  unsigned int baseIdx,
                                                  const unsigned int* __restrict__ cut,
                                                  unsigned int* __restrict__ cnt,
                                                  unsigned long long* __restrict__ cand) {
  int i = blockIdx.x * THRD + threadIdx.x;
  if (i >= n) return;
  float v = s[i];
  if (v < kThr) return;
  int b = (int)((v - kThr) * ((float)kBins / kThr));
  b = min(max(b, 0), kBins - 1);
  if ((unsigned)b < *cut) return;
  unsigned int pos = atomicAdd(cnt, 1u);
  if (pos < (unsigned)kCap) {
    unsigned int sb = __float_as_uint(v);
    cand[pos] = ((unsigned long long)sb << 32) |
                (unsigned long long)(0xFFFFFFFFu - (baseIdx + (unsigned)i));
  }
}

// ---- kernel 4: one-block bitonic sort of 4096 u64 keys, descending ----
__global__ __launch_bounds__(1024) void k_sort(unsigned long long* __restrict__ cand) {
  __shared__ unsigned long long a[kCap];   // 32 KB LDS
  for (int i = threadIdx.x; i < kCap; i += 1024) a[i] = cand[i];
  __syncthreads();
  for (int size = 2; size <= kCap; size <<= 1) {
    for (int stride = size >> 1; stride > 0; stride >>= 1) {
      for (int t = threadIdx.x; t < kCap / 2; t += 1024) {
        const int i = ((t & ~(stride - 1)) << 1) | (t & (stride - 1));
        const int j = i + stride;
        const bool desc = ((i & size) == 0);
        unsigned long long x = a[i], y = a[j];
        if ((x < y) == desc) { a[i] = y; a[j] = x; }
      }
      __syncthreads();
    }
  }
  for (int i = threadIdx.x; i < kCap; i += 1024) cand[i] = a[i];
}

// ---- kernel 5: decode top-K boxes/landmarks ----
__global__ __launch_bounds__(THRD) void k_decode(
    const unsigned long long* __restrict__ cand,
    const float* __restrict__ bb8,  const float* __restrict__ lm8,
    const float* __restrict__ bb16, const float* __restrict__ lm16,
    const float* __restrict__ bb32, const float* __restrict__ lm32,
    const float* __restrict__ detScale,
    float* __restrict__ dec, unsigned int* __restrict__ validArr) {
  int i = blockIdx.x * THRD + threadIdx.x;
  if (i >= kK) return;
  const unsigned long long key = cand[i];
  const float conf = __uint_as_float((unsigned int)(key >> 32));
  const unsigned int gidx = 0xFFFFFFFFu - (unsigned int)(key & 0xFFFFFFFFull);
  const bool valid = (key != 0ull) && (conf >= kThr) && (gidx < (unsigned)kNTot);

  float o[15];
  float area = 0.0f;
  if (valid) {
    unsigned int local; float stride; unsigned int n;
    const float* bb; const float* lm;
    if (gidx < (unsigned)kN8)               { local = gidx;                    stride =  8.f; n = 640; bb = bb8;  lm = lm8;  }
    else if (gidx < (unsigned)(kN8 + kN16)) { local = gidx - kN8;              stride = 16.f; n = 320; bb = bb16; lm = lm16; }
    else                                    { local = gidx - (kN8 + kN16);     stride = 32.f; n = 160; bb = bb32; lm = lm32; }
    const unsigned int cell = local >> 1;
    const float cx = (float)(cell % n) * stride;
    const float cy = (float)(cell / n) * stride;
    const float ds = detScale[0];
    const float* b = bb + (size_t)local * 4;
    const float* l = lm + (size_t)local * 10;
    o[0] = conf;
    o[1] = (cx - b[0] * stride) / ds;
    o[2] = (cy - b[1] * stride) / ds;
    o[3] = (cx + b[2] * stride) / ds;
    o[4] = (cy + b[3] * stride) / ds;
    #pragma unroll
    for (int p = 0; p < 5; ++p) {
      o[5 + 2 * p]     = (cx + l[2 * p]     * stride) / ds;
      o[5 + 2 * p + 1] = (cy + l[2 * p + 1] * stride) / ds;
    }
    area = fmaxf(o[3] - o[1], 0.0f) * fmaxf(o[4] - o[2], 0.0f);
  } else {
    #pragma unroll
    for (int c = 0; c < 15; ++c) o[c] = 0.0f;
  }
  #pragma unroll
  for (int c = 0; c < 15; ++c) dec[(size_t)i * 16 + c] = o[c];
  dec[(size_t)i * 16 + 15] = area;
  validArr[i] = valid ? 1u : 0u;
}

// ---- kernel 6: suppression bitmatrix (K rows x 64 words) ----
__global__ __launch_bounds__(THRD) void k_supp(const float* __restrict__ dec,
                                               unsigned int* __restrict__ supp) {
  int t = blockIdx.x * THRD + threadIdx.x;
  if (t >= kK * (kK / 32)) return;
  const int i = t >> 6;     // row
  const int w = t & 63;     // word
  const float* di = dec + (size_t)i * 16;
  const float x0 = di[1], y0 = di[2], x1 = di[3], y1 = di[4], ai = di[15];
  unsigned int word = 0u;
  const int jb = w * 32;
  #pragma unroll 4
  for (int jj = 0; jj < 32; ++jj) {
    const float* dj = dec + (size_t)(jb + jj) * 16;
    const float ix0 = fmaxf(x0, dj[1]);
    const float iy0 = fmaxf(y0, dj[2]);
    const float ix1 = fminf(x1, dj[3]);
    const float iy1 = fminf(y1, dj[4]);
    const float inter = fmaxf(ix1 - ix0, 0.0f) * fmaxf(iy1 - iy0, 0.0f);
    const float iou = inter / (ai + dj[15] - inter + 1e-9f);
    word |= (iou > kNmsT) ? (1u << jj) : 0u;
  }
  supp[t] = word;
}

// ---- kernel 7: serial greedy NMS reduction (single block, 64 word-owners) ----
__global__ __launch_bounds__(64) void k_nms(const unsigned int* __restrict__ supp,
                                            const unsigned int* __restrict__ validArr,
                                            unsigned int* __restrict__ keepw) {
  __shared__ unsigned int remv[64];
  __shared__ unsigned int kw[64];
  const int tid = threadIdx.x;
  remv[tid] = 0u;
  kw[tid] = 0u;
  __syncthreads();
  for (int i = 0; i < kK; ++i) {
    const int w = i >> 5, b = i & 31;
    const unsigned int rw = remv[w];
    const bool alive = (((rw >> b) & 1u) == 0u) && (validArr[i] != 0u);
    __syncthreads();
    if (alive) {
      unsigned int m = supp[(size_t)i * 64 + tid];
      if (tid < w)       m = 0u;
      else if (tid == w) m &= (b == 31) ? 0u : ~((1u << (b + 1)) - 1u);  // only j > i
      remv[tid] |= m;
      if (tid == w) kw[w] |= (1u << b);
    }
    __syncthreads();
  }
  keepw[tid] = kw[tid];
}

// ---- kernel 8: masked output (K x 15) ----
__global__ __launch_bounds__(THRD) void k_out(const float* __restrict__ dec,
                                              const unsigned int* __restrict__ keepw,
                                              float* __restrict__ out) {
  int t = blockIdx.x * THRD + threadIdx.x;
  if (t >= kK * 15) return;
  const int i = t / 15, c = t % 15;
  const bool kp = (keepw[i >> 5] >> (i & 31)) & 1u;
  out[t] = kp ? dec[(size_t)i * 16 + c] : 0.0f;
}

static inline int cdiv(int a, int b) { return (a + b - 1) / b; }

extern "C" void kernel_launch(void* const* d_in, const int* in_sizes, int n_in,
                              void* d_out, int out_size, void* d_ws, size_t ws_size,
                              hipStream_t stream) {
  const float* s8  = (const float*)d_in[0];
  const float* b8  = (const float*)d_in[1];
  const float* l8  = (const float*)d_in[2];
  const float* s16 = (const float*)d_in[3];
  const float* b16 = (const float*)d_in[4];
  const float* l16 = (const float*)d_in[5];
  const float* s32 = (const float*)d_in[6];
  const float* b32 = (const float*)d_in[7];
  const float* l32 = (const float*)d_in[8];
  const float* dsc = (const float*)d_in[9];
  float* out = (float*)d_out;

  char* ws = (char*)d_ws;
  unsigned int*       hist = (unsigned int*)(ws + WS_HIST);
  unsigned int*       cnt  = (unsigned int*)(ws + WS_CNT);
  unsigned int*       cut  = (unsigned int*)(ws + WS_CUT);
  unsigned long long* cand = (unsigned long long*)(ws + WS_CAND);
  float*              dec  = (float*)(ws + WS_DEC);
  unsigned int*       val  = (unsigned int*)(ws + WS_VAL);
  unsigned int*       kw   = (unsigned int*)(ws + WS_KEEP);
  unsigned int*       supp = (unsigned int*)(ws + WS_SUPP);

  k_init<<<1, 1024, 0, stream>>>(hist, cnt, cand);

  // ~4 tiles per block keeps the async double-buffer pipeline busy.
  auto histGrid = [](int n) { int g = cdiv(cdiv(n, kTile), 4); return g < 1 ? 1 : g; };
  k_hist<<<histGrid(kN8),  THRD, 0, stream>>>(s8,  kN8,  hist);
  k_hist<<<histGrid(kN16), THRD, 0, stream>>>(s16, kN16, hist);
  k_hist<<<histGrid(kN32), THRD, 0, stream>>>(s32, kN32, hist);

  k_scan<<<1, 32, 0, stream>>>(hist, cut);

  k_collect<<<cdiv(kN8,  THRD), THRD, 0, stream>>>(s8,  kN8,  0u,                      cut, cnt, cand);
  k_collect<<<cdiv(kN16, THRD), THRD, 0, stream>>>(s16, kN16, (unsigned)kN8,           cut, cnt, cand);
  k_collect<<<cdiv(kN32, THRD), THRD, 0, stream>>>(s32, kN32, (unsigned)(kN8 + kN16),  cut, cnt, cand);

  k_sort<<<1, 1024, 0, stream>>>(cand);

  k_decode<<<cdiv(kK, THRD), THRD, 0, stream>>>(cand, b8, l8, b16, l16, b32, l32, dsc, dec, val);
  k_supp<<<cdiv(kK * 64, THRD), THRD, 0, stream>>>(dec, supp);
  k_nms<<<1, 64, 0, stream>>>(supp, val, kw);
  k_out<<<cdiv(kK * 15, THRD), THRD, 0, stream>>>(dec, kw, out);
}